// LengthRegulator_7645041787357
// MI455X (gfx1250) — compile-verified
//
#include <hip/hip_runtime.h>
#include <hip/hip_bf16.h>
#include <stdint.h>

#define GLOBAL_AS __attribute__((address_space(1)))
#define LDS_AS    __attribute__((address_space(3)))

// Vector type matching the async-LDS builtins' expected parameter type.
typedef int v4i __attribute__((vector_size(16)));

// Cache-policy (CPol) temporal hints, gfx12+ encoding: TH in bits [2:0].
#define CPOL_RT 0
#define CPOL_NT 1

// ---------------------------------------------------------------------------
// CDNA5 async global<->LDS copies (ASYNCcnt-tracked). Builtins verified to
// exist on this toolchain: (v4i AS1*, v4i AS3*, Ii offset, Ii cpol).
// ---------------------------------------------------------------------------
__device__ __forceinline__ void async_load_b128(const void* gsrc, void* ldst) {
#if defined(__HIP_DEVICE_COMPILE__)
#if __has_builtin(__builtin_amdgcn_global_load_async_to_lds_b128)
    __builtin_amdgcn_global_load_async_to_lds_b128(
        (GLOBAL_AS v4i*)(uintptr_t)gsrc,
        (LDS_AS v4i*)(uint32_t)(uintptr_t)ldst,
        0, CPOL_RT);
#else
    unsigned l = (unsigned)(uintptr_t)ldst;          // LDS byte offset (low 32 bits)
    unsigned long long g = (unsigned long long)(uintptr_t)gsrc;
    asm volatile("global_load_async_to_lds_b128 %0, %1, off"
                 :: "v"(l), "v"(g) : "memory");
#endif
#endif
}

__device__ __forceinline__ void async_store_b128_nt(void* gdst, const void* lsrc) {
#if defined(__HIP_DEVICE_COMPILE__)
#if __has_builtin(__builtin_amdgcn_global_store_async_from_lds_b128)
    __builtin_amdgcn_global_store_async_from_lds_b128(
        (GLOBAL_AS v4i*)(uintptr_t)gdst,
        (LDS_AS v4i*)(uint32_t)(uintptr_t)lsrc,
        0, CPOL_NT);
#else
    unsigned l = (unsigned)(uintptr_t)lsrc;
    unsigned long long g = (unsigned long long)(uintptr_t)gdst;
    asm volatile("global_store_async_from_lds_b128 %0, %1, off th:TH_STORE_NT"
                 :: "v"(g), "v"(l) : "memory");
#endif
#endif
}

__device__ __forceinline__ void wait_async0() {
#if defined(__HIP_DEVICE_COMPILE__)
#if __has_builtin(__builtin_amdgcn_s_wait_asynccnt)
    __builtin_amdgcn_s_wait_asynccnt(0);
#else
    asm volatile("s_wait_asynccnt 0x0" ::: "memory");
#endif
#endif
}

// ---------------------------------------------------------------------------
// Problem constants (fixed by the reference setup).
// ---------------------------------------------------------------------------
static constexpr int Bc = 32;
static constexpr int Sc = 1024;
static constexpr int Dc = 512;

// ---------------------------------------------------------------------------
// Kernel 2 (placed first so the disasm snippet shows it): scatter-by-source-
// token expansion. One 128-thread block per (b, s). The 2KB row is async-
// DMA'd global->LDS exactly once, then fanned out `duration` times LDS->
// global with non-temporal async stores. Each lane's LDS bytes are private
// to it, so only the per-wave ASYNCcnt wait is needed (no block barrier).
// ---------------------------------------------------------------------------
__global__ void lr_expand_kernel(const float* __restrict__ x,
                                 const int* __restrict__ csum,
                                 float* __restrict__ out,
                                 int max_len) {
    const int s   = blockIdx.x;   // source token
    const int b   = blockIdx.y;   // batch
    const int tid = threadIdx.x;  // 0..127, each lane moves one float4 (16B)

    const int end   = csum[b * Sc + s];
    const int start = (s == 0) ? 0 : csum[b * Sc + s - 1];
    const int d     = end - start;    // duration, 1..8

    __shared__ __align__(16) float row[Dc];

    const float* src = x + ((size_t)b * Sc + s) * Dc + tid * 4;
    async_load_b128(src, &row[tid * 4]);
    wait_async0();  // LDS data valid for this wave's lanes

    float* dst = out + ((size_t)b * max_len + start) * Dc + tid * 4;
    for (int k = 0; k < d; ++k) {
        async_store_b128_nt(dst + (size_t)k * Dc, &row[tid * 4]);
    }
    wait_async0();
}

// ---------------------------------------------------------------------------
// Kernel 1: per-batch inclusive scan of durations -> csum; emit mel_len.
// One 1024-thread block per batch. Runs once; cost is negligible.
// ---------------------------------------------------------------------------
__global__ void lr_scan_kernel(const int* __restrict__ dur,
                               int* __restrict__ csum,
                               int* __restrict__ mel_len_i,
                               float* __restrict__ mel_len_out) {
    __shared__ int buf[Sc];
    const int b = blockIdx.x;
    const int t = threadIdx.x;
    buf[t] = dur[b * Sc + t];
    __syncthreads();
    // Hillis-Steele inclusive scan over 1024 elements.
    for (int off = 1; off < Sc; off <<= 1) {
        int v = (t >= off) ? buf[t - off] : 0;
        __syncthreads();
        buf[t] += v;
        __syncthreads();
    }
    csum[b * Sc + t] = buf[t];
    if (t == Sc - 1) {
        mel_len_i[b]   = buf[t];
        mel_len_out[b] = (float)buf[t];
    }
}

// ---------------------------------------------------------------------------
// Kernel 3: mel_mask + zero-fill of tail rows (t >= mel_len[b]).
// One 128-thread block per (b, t): thread 0 writes the mask element; if the
// row is masked, all 128 threads zero the 512-float row with float4 stores.
// ---------------------------------------------------------------------------
__global__ void lr_mask_kernel(const int* __restrict__ mel_len_i,
                               float* __restrict__ out,
                               float* __restrict__ mask_out,
                               int max_len) {
    const int t   = blockIdx.x;
    const int b   = blockIdx.y;
    const int tid = threadIdx.x;

    const int  ml     = mel_len_i[b];
    const bool masked = (t >= ml);

    if (tid == 0) {
        mask_out[(size_t)b * max_len + t] = masked ? 1.0f : 0.0f;
    }
    if (masked) {
        float4 z = make_float4(0.f, 0.f, 0.f, 0.f);
        float* row = out + ((size_t)b * max_len + t) * Dc;
        *(float4*)(row + tid * 4) = z;
    }
}

// ---------------------------------------------------------------------------
// Launch. Output layout (flattened tuple, all float):
//   [0, B*max_len*D)                       expanded
//   [B*max_len*D, B*max_len*D + B)         mel_len (as float)
//   [.. + B, .. + B + B*max_len)           mel_mask (0.0 / 1.0)
// max_len recovered from out_size = B + B*max_len*(D+1).
// ---------------------------------------------------------------------------
extern "C" void kernel_launch(void* const* d_in, const int* in_sizes, int n_in,
                              void* d_out, int out_size, void* d_ws, size_t ws_size,
                              hipStream_t stream) {
    (void)in_sizes; (void)n_in; (void)ws_size;

    const float* x   = (const float*)d_in[0];
    const int*   dur = (const int*)d_in[1];
    float*       out = (float*)d_out;

    const int max_len = (out_size - Bc) / (Bc * (Dc + 1));

    // Workspace: csum (B*S ints) then mel_len (B ints) -> 131,200 bytes.
    int* csum   = (int*)d_ws;
    int* mlen_i = csum + Bc * Sc;

    float* mel_len_out = out + (size_t)Bc * max_len * Dc;
    float* mask_out    = mel_len_out + Bc;

    lr_scan_kernel<<<dim3(Bc), dim3(Sc), 0, stream>>>(dur, csum, mlen_i, mel_len_out);
    lr_expand_kernel<<<dim3(Sc, Bc), dim3(128), 0, stream>>>(x, csum, out, max_len);
    lr_mask_kernel<<<dim3((unsigned)max_len, Bc), dim3(128), 0, stream>>>(mlen_i, out, mask_out, max_len);
}